// DynamicWeights_635655160218
// MI455X (gfx1250) — compile-verified
//
#include <hip/hip_runtime.h>
#include <cstdint>

#define NB 8
#define CC 640
#define HH 48
#define WW 48
#define KK 9
#define HW (HH * WW)
#define CCHUNKS 4
#define CPER (CC / CCHUNKS)   // 160 channels per block
#define TPB 192               // 6 waves of 32: 16 channel-lanes x 12 w-quads

__global__ __launch_bounds__(TPB)
void DynamicWeights_main_kernel(const float* __restrict__ q,
                                const float* __restrict__ df,
                                float* __restrict__ out)
{
    __shared__ float sfilt[WW * KK];   // 432 floats: this row's per-pixel 3x3 filters

    const int bx = blockIdx.x;         // n*H + h
    const int n  = bx / HH;
    const int h  = bx % HH;
    const int c0 = blockIdx.y * CPER;
    const int tid = threadIdx.x;

    // ---- Stage dynamic_filter row into LDS via gfx1250 async global->LDS DMA ----
    // df layout [N, H*W, 9, 1]: row slice is 432 contiguous floats.
    const float* frow = df + ((size_t)n * HW + (size_t)h * WW) * KK;
    for (int idx = tid; idx < WW * KK; idx += TPB) {
        uint32_t lds_addr = (uint32_t)(uintptr_t)(&sfilt[idx]);
        const float* gp = frow + idx;
        asm volatile("global_load_async_to_lds_b32 %0, %1, off"
                     :: "v"(lds_addr), "v"(gp)
                     : "memory");
    }
    asm volatile("s_wait_asynccnt 0" ::: "memory");
    __syncthreads();

    const int wq = tid % 12;           // which float4 along the row
    const int cl = tid / 12;           // channel lane 0..15
    const int w0 = wq * 4;

    // Hoist this thread's 36 filter taps (4 pixels x 9 taps) into registers:
    // tap k = dy*3 + dx multiplies q[h+dy-1][w+dx-1].
    float ff[3][3][4];
    #pragma unroll
    for (int i = 0; i < 4; ++i)
        #pragma unroll
        for (int t = 0; t < 9; ++t)
            ff[t / 3][t % 3][i] = sfilt[(w0 + i) * KK + t];

    const float* qbase = q   + ((size_t)n * CC + c0) * HW;
    float*       obase = out + ((size_t)n * CC + c0) * HW;

    const bool r0ok = (h > 0);
    const bool r2ok = (h < HH - 1);

    for (int c = cl; c < CPER; c += 16) {
        const float* qc = qbase + (size_t)c * HW;

        if (c + 16 < CPER)  // gfx1250 global_prefetch of next channel's center row
            __builtin_prefetch(qbase + (size_t)(c + 16) * HW + (size_t)h * WW + w0, 0, 1);

        float a0 = 0.f, a1 = 0.f, a2 = 0.f, a3 = 0.f;
        #pragma unroll
        for (int dy = 0; dy < 3; ++dy) {
            const int row = h + dy - 1;
            const bool ok = (dy == 0) ? r0ok : (dy == 2 ? r2ok : true); // uniform branch
            float x0 = 0.f, x1 = 0.f, x2 = 0.f, x3 = 0.f, x4 = 0.f, x5 = 0.f;
            if (ok) {
                const float* qr = qc + (size_t)row * WW;
                const float4 m = *(const float4*)(qr + w0);   // 16B-aligned (row=192B)
                x1 = m.x; x2 = m.y; x3 = m.z; x4 = m.w;
                x0 = (wq > 0)  ? qr[w0 - 1] : 0.f;            // left halo
                x5 = (wq < 11) ? qr[w0 + 4] : 0.f;            // right halo
            }
            a0 = fmaf(ff[dy][0][0], x0, fmaf(ff[dy][1][0], x1, fmaf(ff[dy][2][0], x2, a0)));
            a1 = fmaf(ff[dy][0][1], x1, fmaf(ff[dy][1][1], x2, fmaf(ff[dy][2][1], x3, a1)));
            a2 = fmaf(ff[dy][0][2], x2, fmaf(ff[dy][1][2], x3, fmaf(ff[dy][2][2], x4, a2)));
            a3 = fmaf(ff[dy][0][3], x3, fmaf(ff[dy][1][3], x4, fmaf(ff[dy][2][3], x5, a3)));
        }
        float4 o;
        o.x = a0 > 0.f ? a0 : 0.f;
        o.y = a1 > 0.f ? a1 : 0.f;
        o.z = a2 > 0.f ? a2 : 0.f;
        o.w = a3 > 0.f ? a3 : 0.f;
        *(float4*)(obase + (size_t)c * HW + (size_t)h * WW + w0) = o;
    }
}

__global__ void DynamicWeights_zero_kernel(float4* __restrict__ p, int n4)
{
    int i = blockIdx.x * blockDim.x + threadIdx.x;
    if (i < n4) p[i] = make_float4(0.f, 0.f, 0.f, 0.f);
}

extern "C" void kernel_launch(void* const* d_in, const int* in_sizes, int n_in,
                              void* d_out, int out_size, void* d_ws, size_t ws_size,
                              hipStream_t stream)
{
    (void)in_sizes; (void)n_in; (void)out_size; (void)d_ws; (void)ws_size;

    const float* q  = (const float*)d_in[0];   // [8, 640, 48, 48] fp32
    const float* df = (const float*)d_in[1];   // [8, 2304, 9, 1]  fp32
    float* out = (float*)d_out;                // out1 (11796480) ++ out2 (165888)

    dim3 grid(NB * HH, CCHUNKS);
    DynamicWeights_main_kernel<<<grid, TPB, 0, stream>>>(q, df, out);

    float* out2 = out + (size_t)NB * CC * HW;  // 16B-aligned offset
    const int n2  = NB * HW * KK;              // 165888
    const int n4  = n2 / 4;                    // 41472 float4 stores
    DynamicWeights_zero_kernel<<<(n4 + 255) / 256, 256, 0, stream>>>((float4*)out2, n4);
}